// AlignedAttention_44246753083730
// MI455X (gfx1250) — compile-verified
//
#include <hip/hip_runtime.h>
#include <hip/hip_bf16.h>
#include <math.h>

// Problem constants (from reference)
#define BB   8
#define PLEN 2048
#define QLEN 1024
#define HID  1024

typedef __attribute__((ext_vector_type(2))) float v2f;
typedef __attribute__((ext_vector_type(8))) float v8f;
typedef int v4i_vs __attribute__((vector_size(16)));   // matches builtin param type

// ---------------------------------------------------------------------------
// CDNA5 async global->LDS copy (GLOBAL_LOAD_ASYNC_TO_LDS_B128, ASYNCcnt).
// Builtin signature (from hipcc diagnostic): 
//   (v4i_vs addrspace(1)*, v4i_vs addrspace(3)*, imm offset, imm cpol)
// Guarded: falls back to a register round-trip if the builtin is absent.
// ---------------------------------------------------------------------------
#if __has_builtin(__builtin_amdgcn_global_load_async_to_lds_b128)
#define HAS_ASYNC_LDS 1
__device__ __forceinline__ void copy16_g2l(float* lds, const float* g) {
    __builtin_amdgcn_global_load_async_to_lds_b128(
        (__attribute__((address_space(1))) v4i_vs*)(g),
        (__attribute__((address_space(3))) v4i_vs*)(lds),
        /*offset=*/0, /*cpol=*/0);
}
template <int N>
__device__ __forceinline__ void wait_async() {
#if __has_builtin(__builtin_amdgcn_s_wait_asynccnt)
    __builtin_amdgcn_s_wait_asynccnt(N);
#else
    asm volatile("s_wait_asynccnt %0" ::"n"(N));
#endif
}
#else
#define HAS_ASYNC_LDS 0
__device__ __forceinline__ void copy16_g2l(float* lds, const float* g) {
    *(float4*)lds = *(const float4*)g;   // synchronous fallback
}
template <int N>
__device__ __forceinline__ void wait_async() {}
#endif

// ---------------------------------------------------------------------------
// Tiled fp32 WMMA GEMM:  C[M,N] = op(A[M,K] x B)  (+ optional ReLU epilogue)
//   BT = true : B stored [N,K] row-major (C = A * B^T)   -- stages 1,2,3
//   BT = false: B stored [K,N] row-major (C = A * B)     -- stage 5
// Workgroup: 256 threads (8 wave32), tile 128(M) x 128(N), K-block 32,
// LDS double-buffered with async global->LDS DMA overlapping WMMA compute.
// Wave tile: 64(M) x 32(N) = 4x2 grid of 16x16 WMMA accumulators.
// Assumes M%128==0, N%128==0, K%32==0 (true for all stages here).
// ---------------------------------------------------------------------------
template <bool BT, bool RELU>
__global__ __launch_bounds__(256)
void gemm_wmma_f32(const float* __restrict__ A, const float* __restrict__ Bm,
                   float* __restrict__ C,
                   long batchStrideA, long batchStrideB, long batchStrideC,
                   int M, int N, int K)
{
    A  += (size_t)blockIdx.z * batchStrideA;
    Bm += (size_t)blockIdx.z * batchStrideB;
    C  += (size_t)blockIdx.z * batchStrideC;

    const int m0   = blockIdx.y * 128;
    const int n0   = blockIdx.x * 128;
    const int t    = threadIdx.x;
    const int lane = t & 31;
    const int wave = t >> 5;
    const int waveM = wave >> 2;       // 0..1  -> 64-row slab
    const int waveN = wave & 3;        // 0..3  -> 32-col slab
    const int r16  = lane & 15;
    const int kh   = (lane >> 4) << 1; // 0 or 2 (K-half per ISA A/B layout)

    // Double-buffered LDS tiles, padded to avoid bank conflicts.
    __shared__ float sA[2][128 * 36];  // [m][k], stride 36
    __shared__ float sB[2][128 * 36];  // BT: [n][k] stride 36 ; !BT: [k][n] stride 132

    // Per-thread stationary addressing for the staging phase.
    const int rA = t >> 3, cA = (t & 7) << 2;     // A & B(BT): 32 rows/pass, 8 f4/row
    const int rN = t >> 5, cN = (t & 31) << 2;    // B(!BT): 8 rows/pass, 32 f4/row

    // Issue one K-block's 8 async b128 copies (4 A + 4 B) for this thread.
    auto issue = [&](int kb, int buf) {
        const int k0 = kb << 5;
#pragma unroll
        for (int it = 0; it < 4; ++it) {
            int r = rA + it * 32;
            copy16_g2l(&sA[buf][r * 36 + cA],
                       A + (size_t)(m0 + r) * K + k0 + cA);
        }
        if (BT) {
#pragma unroll
            for (int it = 0; it < 4; ++it) {
                int r = rA + it * 32;
                copy16_g2l(&sB[buf][r * 36 + cA],
                           Bm + (size_t)(n0 + r) * K + k0 + cA);
            }
        } else {
#pragma unroll
            for (int it = 0; it < 4; ++it) {
                int r = rN + it * 8;              // K rows 0..31
                copy16_g2l(&sB[buf][r * 132 + cN],
                           Bm + (size_t)(k0 + r) * N + n0 + cN);
            }
        }
    };

    v8f acc[4][2] = {}; // zero accumulators

    const int nKB = K >> 5;
    issue(0, 0);
    for (int kb = 0; kb < nKB; ++kb) {
        const int buf = kb & 1;
        if (kb + 1 < nKB) {
            issue(kb + 1, buf ^ 1);   // prefetch next block into other buffer
            wait_async<8>();          // drain previous batch only (in-order)
        } else {
            wait_async<0>();
        }
        __syncthreads();

        // ---- compute: 8 steps x 8 WMMA ----
#pragma unroll
        for (int kk = 0; kk < 32; kk += 4) {
            v2f aF[4], bF[2];
#pragma unroll
            for (int i = 0; i < 4; ++i)
                aF[i] = *(const v2f*)&sA[buf][(waveM * 64 + i * 16 + r16) * 36 + kk + kh];
            if (BT) {
#pragma unroll
                for (int j = 0; j < 2; ++j)
                    bF[j] = *(const v2f*)&sB[buf][(waveN * 32 + j * 16 + r16) * 36 + kk + kh];
            } else {
#pragma unroll
                for (int j = 0; j < 2; ++j) {
                    v2f b;
                    b.x = sB[buf][(kk + kh)     * 132 + waveN * 32 + j * 16 + r16];
                    b.y = sB[buf][(kk + kh + 1) * 132 + waveN * 32 + j * 16 + r16];
                    bF[j] = b;
                }
            }
#pragma unroll
            for (int i = 0; i < 4; ++i)
#pragma unroll
                for (int j = 0; j < 2; ++j)
                    acc[i][j] = __builtin_amdgcn_wmma_f32_16x16x4_f32(
                        /*neg_a=*/false, aF[i], /*neg_b=*/false, bF[j],
                        /*c_mod=*/(short)0, acc[i][j],
                        /*reuse_a=*/false, /*reuse_b=*/false);
        }
        __syncthreads();   // protect buf before it is overwritten next round
    }

    // ---- epilogue: C/D layout VGPR v -> row v + 8*(lane>=16), col lane&15 ----
    const int mBase = m0 + waveM * 64;
    const int nBase = n0 + waveN * 32;
    const int mHalf = (lane >> 4) << 3;
#pragma unroll
    for (int i = 0; i < 4; ++i)
#pragma unroll
        for (int j = 0; j < 2; ++j)
#pragma unroll
            for (int v = 0; v < 8; ++v) {
                int mm = mBase + i * 16 + v + mHalf;
                int nn = nBase + j * 16 + r16;
                float x = acc[i][j][v];
                if (RELU) x = fmaxf(x, 0.0f);
                C[(size_t)mm * N + nn] = x;
            }
}

// ---------------------------------------------------------------------------
// Row-wise masked softmax over Q=1024, in place. One 256-thread block per row.
// ---------------------------------------------------------------------------
__global__ __launch_bounds__(256)
void softmax_rows(float* __restrict__ scores, const unsigned char* __restrict__ qmask,
                  int Q, int P)
{
    const int row = blockIdx.x;             // 0 .. B*P-1
    const int b   = row / P;
    float* s = scores + (size_t)row * Q;
    const unsigned char* m = qmask + (size_t)b * Q;
    const int t = threadIdx.x;

    __shared__ float red[256];

    float vals[4];
    float mx = -INFINITY;
#pragma unroll
    for (int i = 0; i < 4; ++i) {
        int qi = t + (i << 8);
        float x = s[qi];
        if (m[qi]) x = -INFINITY;           // q_mask true => pad => -inf
        vals[i] = x;
        mx = fmaxf(mx, x);
    }
    red[t] = mx; __syncthreads();
    for (int off = 128; off > 0; off >>= 1) {
        if (t < off) red[t] = fmaxf(red[t], red[t + off]);
        __syncthreads();
    }
    mx = red[0]; __syncthreads();

    float sum = 0.0f;
#pragma unroll
    for (int i = 0; i < 4; ++i) {
        float e = expf(vals[i] - mx);       // exp(-inf) = 0 for masked entries
        vals[i] = e;
        sum += e;
    }
    red[t] = sum; __syncthreads();
    for (int off = 128; off > 0; off >>= 1) {
        if (t < off) red[t] += red[t + off];
        __syncthreads();
    }
    const float inv = 1.0f / red[0];
#pragma unroll
    for (int i = 0; i < 4; ++i) s[t + (i << 8)] = vals[i] * inv;
}

// ---------------------------------------------------------------------------
// Harness entry point.
//   d_in : [0]=k [B,P,H] f32, [1]=q [B,Q,H] f32, [2]=q_mask [B,Q] bool(1B),
//          [3]=Wk [H,H] f32, [4]=Wq [H,H] f32
//   d_out: ctx [B,P,H] f32  ||  alphas [B,P,Q] f32   (concatenated)
//   d_ws : p_key (B*P*H f32, 64 MiB) || q_key (B*Q*H f32, 32 MiB)
// ---------------------------------------------------------------------------
extern "C" void kernel_launch(void* const* d_in, const int* in_sizes, int n_in,
                              void* d_out, int out_size, void* d_ws, size_t ws_size,
                              hipStream_t stream)
{
    (void)in_sizes; (void)n_in; (void)out_size; (void)ws_size;

    const float*         k     = (const float*)d_in[0];
    const float*         q     = (const float*)d_in[1];
    const unsigned char* qmask = (const unsigned char*)d_in[2];
    const float*         Wk    = (const float*)d_in[3];
    const float*         Wq    = (const float*)d_in[4];

    float* ctx    = (float*)d_out;
    float* alphas = ctx + (size_t)BB * PLEN * HID;

    float* p_key = (float*)d_ws;                          // [B,P,H]
    float* q_key = p_key + (size_t)BB * PLEN * HID;       // [B,Q,H]

    const dim3 blk(256);

    // 1) p_key = relu(k @ Wk^T)   (Wk batch-broadcast)
    gemm_wmma_f32<true, true><<<dim3(HID / 128, PLEN / 128, BB), blk, 0, stream>>>(
        k, Wk, p_key, (long)PLEN * HID, 0L, (long)PLEN * HID, PLEN, HID, HID);

    // 2) q_key = relu(q @ Wq^T)
    gemm_wmma_f32<true, true><<<dim3(HID / 128, QLEN / 128, BB), blk, 0, stream>>>(
        q, Wq, q_key, (long)QLEN * HID, 0L, (long)QLEN * HID, QLEN, HID, HID);

    // 3) scores = p_key @ q_key^T  -> written into the alphas slot of d_out
    gemm_wmma_f32<true, false><<<dim3(QLEN / 128, PLEN / 128, BB), blk, 0, stream>>>(
        p_key, q_key, alphas, (long)PLEN * HID, (long)QLEN * HID, (long)PLEN * QLEN,
        PLEN, QLEN, HID);

    // 4) alphas = softmax(mask(scores)) in place, row-wise over Q
    softmax_rows<<<dim3(BB * PLEN), blk, 0, stream>>>(alphas, qmask, QLEN, PLEN);

    // 5) ctx = alphas @ q   (B = q stored [Q, HID] row-major -> BT=false)
    gemm_wmma_f32<false, false><<<dim3(HID / 128, PLEN / 128, BB), blk, 0, stream>>>(
        alphas, q, ctx, (long)PLEN * QLEN, (long)QLEN * HID, (long)PLEN * HID,
        PLEN, HID, QLEN);
}